// SAvgPool2d_69595650064851
// MI455X (gfx1250) — compile-verified
//
#include <hip/hip_runtime.h>
#include <hip/hip_bf16.h>
#include <stdint.h>

// SNN avg-pool(2x2) + leaky integrate-and-fire over T.
// x: [B=16,C=64,H=64,W=64,T=32] f32 (T contiguous) -> out: [B,C,32,32,32] f32
//
// Pixel-group g=(b,c,hp) consumes exactly global bytes [g*16KB,(g+1)*16KB)
// (rows 2hp, 2hp+1). One wave32 block owns one group: a single TDM
// descriptor (inline asm: portable across ROCm7.2/clang-22 and
// amdgpu-toolchain/clang-23, whose tensor builtins differ in arity) DMAs
// the 16KB window into LDS with hardware padding of 16B per 256B (kills
// the 256B-stride bank conflicts of the pooling reads). The wave waits on
// TENSORcnt, pools, runs the serial T-recurrence in registers, and
// streams NT stores out. No barriers: single-wave workgroups.

typedef float        v4f   __attribute__((ext_vector_type(4)));
typedef unsigned int u32x4 __attribute__((ext_vector_type(4)));
typedef unsigned int u32x8 __attribute__((ext_vector_type(8)));

// Flip to 0 to fall back to the per-lane async copy path.
#define USE_TDM_ASM 1

#define TPB 32
#define GROUP_CHUNKS  1024                         // 16KB window in 16B chunks
#define PAD_CHUNKS    (GROUP_CHUNKS / 16)          // one 16B pad per 256B
#define LDS_CHUNKS    (GROUP_CHUNKS + PAD_CHUNKS)  // 1088 -> 17408 B

__global__ __launch_bounds__(TPB) void snn_pool_lif_kernel(
    const float* __restrict__ x,
    const float* __restrict__ thresh,
    float* __restrict__ out)
{
  __shared__ v4f tile[LDS_CHUNKS];             // 17408 B, padded layout

  const uint32_t wp = threadIdx.x;             // output wp, 0..31
  // Low 32 bits of a flat LDS pointer == LDS byte offset (ISA 10.2).
  const uint32_t lds_base = (uint32_t)(uintptr_t)(&tile[0]);
  const float    th   = thresh[0];             // uniform s_load, overlaps DMA
  const uint64_t gsrc = (uint64_t)(uintptr_t)x +
                        (uint64_t)blockIdx.x * (GROUP_CHUNKS * 16u);

#if USE_TDM_ASM
  {
    // ---- D# group 0 (ISA 8.3): count=1 | lds_addr | global_addr | type=2 ----
    u32x4 g0;
    g0[0] = 1u;                                   // count=1, user, no gather
    g0[1] = lds_base;                             // lds_addr (bytes)
    g0[2] = (uint32_t)gsrc;                       // global_addr[31:0]
    g0[3] = (uint32_t)(gsrc >> 32) | (2u << 30);  // global_addr[56:32]|type=2

    // ---- D# group 1 (ISA 8.4): 1-D tile of 4096 f32 with LDS padding ----
    u32x8 g1;
    g1[0] = (2u << 16)           // data_size = 4B
          | (1u << 20)           // pad_enable: pad LDS dest
          | (5u << 22)           // pad_interval code 5 = 64 DW = 256B
          | (3u << 25);          // pad_amount  code 3 = 4 DW  = 16B
    g1[1] = 4096u << 16;         // atomic_bar_addr=0 | tensor_dim0.lo = 4096
    g1[2] = 1u << 16;            // tensor_dim0.hi=0  | tensor_dim1.lo = 1
    g1[3] = 4096u << 16;         // tensor_dim1.hi=0  | tile_dim0 = 4096
    g1[4] = 1u;                  // tile_dim1=1 | tile_dim2=0
    g1[5] = 4096u;               // tensor_dim0_stride[31:0]
    g1[6] = 4096u << 16;         // stride0[47:32]=0 | tensor_dim1_stride.lo
    g1[7] = 0u;                  // tensor_dim1_stride.hi

    // VIMAGE tensor op: VADDR0 = 4-SGPR group0, VADDR1 = 8-SGPR group1,
    // VADDR2/VADDR3 = NULL (<=2-D tensor). One issue per wave.
    asm volatile("tensor_load_to_lds %0, %1, null, null"
                 :: "s"(g0), "s"(g1) : "memory");
    __builtin_amdgcn_s_wait_tensorcnt((short)0);
  }
#else
  // Fallback: per-lane coalesced async copies into the same padded layout.
#pragma unroll
  for (int i = 0; i < GROUP_CHUNKS / TPB; ++i) {     // 32 chunks/lane
    uint32_t n   = (uint32_t)i * TPB + wp;           // linear 16B chunk
    uint32_t off = lds_base + (n + (n >> 4)) * 16u;  // +16B pad per 256B
    uint64_t ga  = gsrc + (uint64_t)n * 16u;
    asm volatile("global_load_async_to_lds_b128 %0, %1, off"
                 :: "v"(off), "v"(ga) : "memory");
  }
  asm volatile("s_wait_asynccnt 0" ::: "memory");
#endif
  __syncthreads();   // compiler fence; S_NOP-class for a 1-wave workgroup

  // ---- 2x2 pool from padded LDS ----
  // Group tile linear layout: [h2(2)][w(64)][k(8 x 16B of t)] chunks,
  // padded chunk index = n + (n>>4). For n = base+k (k<8, base%16 in {0,8})
  // the pad offset is constant across k -> contiguous 8-chunk reads.
  // Lane-to-lane stride = 17 chunks = 272B -> <=2-way bank conflicts.
  v4f acc[8];
#pragma unroll
  for (int k = 0; k < 8; ++k) { v4f z = {0.f, 0.f, 0.f, 0.f}; acc[k] = z; }

#pragma unroll
  for (int c = 0; c < 4; ++c) {
    const uint32_t h2   = (uint32_t)c >> 1;
    const uint32_t par  = (uint32_t)c & 1u;
    const uint32_t base = h2 * 512u + (2u * wp + par) * 8u;
    const uint32_t pb   = base + (base >> 4);        // padded chunk index
#pragma unroll
    for (int k = 0; k < 8; ++k) {
      acc[k] += tile[pb + (uint32_t)k];              // ds_load_b128
    }
  }

  // ---- leaky integrate-and-fire: u_t = u - s*th + p_t ; s_t = (u_t > 0) ----
  float u = 0.f, s = 0.f;
  v4f o[8];
#pragma unroll
  for (int k = 0; k < 8; ++k) {
#pragma unroll
    for (int j = 0; j < 4; ++j) {
      u = u - s * th + 0.25f * acc[k][j];
      s = (u > 0.f) ? 1.f : 0.f;
      o[k][j] = s;
    }
  }

  // ---- NT streamed store (write-once output) ----
  float* dst = out + ((size_t)blockIdx.x * TPB + wp) * 32u;
#pragma unroll
  for (int k = 0; k < 8; ++k) {
    __builtin_nontemporal_store(o[k], (v4f*)(dst + 4 * k));
  }
}

extern "C" void kernel_launch(void* const* d_in, const int* in_sizes, int n_in,
                              void* d_out, int out_size, void* d_ws, size_t ws_size,
                              hipStream_t stream) {
  const float* x      = (const float*)d_in[0];   // 16*64*64*64*32 floats
  const float* thresh = (const float*)d_in[1];   // 1 float
  float* out          = (float*)d_out;           // 16*64*32*32*32 floats

  const int n_pixels = 16 * 64 * 32 * 32;        // 1048576
  dim3 grid(n_pixels / TPB);                     // 32768 groups
  dim3 block(TPB);                               // one wave32 per group
  hipLaunchKernelGGL(snn_pool_lif_kernel, grid, block, 0, stream,
                     x, thresh, out);
}